// ProductVQ_46222438039689
// MI455X (gfx1250) — compile-verified
//
#include <hip/hip_runtime.h>

typedef __attribute__((ext_vector_type(2))) float v2f;
typedef __attribute__((ext_vector_type(8))) float v8f;

#define WPB 4     // waves per block (wave32)
#define MROWS 32  // rows per wave (2 x 16-row C tiles sharing B fragments)

__global__ void zero_loss_kernel(float* loss) { *loss = 0.0f; }

// One wave per codebook row: wnorm[k] = ||W_k||^2 (precomputed so the VQ hot
// loop is pure loads + f32 WMMA; f32 WMMA does not co-execute with VALU).
__global__ __launch_bounds__(32) void wnorm_kernel(const float* __restrict__ W,
                                                   float* __restrict__ out,
                                                   int D) {
  const int row = blockIdx.x;
  const float* wr = W + (size_t)row * D;
  float s = 0.0f;
  for (int d = threadIdx.x; d < D; d += 32) {
    const float v = wr[d];
    s += v * v;
  }
#pragma unroll
  for (int mask = 1; mask < 32; mask <<= 1) s += __shfl_xor(s, mask);
  if (threadIdx.x == 0) out[row] = s;
}

// One wave computes a 32-row tile: scores via V_WMMA_F32_16X16X4_F32 (two
// accumulators share each B fragment), running argmin over K, gather
// q = W[idx], commitment-loss accumulation.
template <int D>
__global__ __launch_bounds__(32 * WPB) void vq_argmin_gather_kernel(
    const float* __restrict__ x,      // (N, D)
    const float* __restrict__ W,      // (K, D)
    const float* __restrict__ wnorm,  // (K,) precomputed ||W_k||^2
    float* __restrict__ q_out,        // (N, D)
    float* __restrict__ idx_out,      // (N,) stored as float
    float* __restrict__ loss_out,     // scalar accumulator
    int K, int N) {
  __shared__ float xs[WPB][MROWS][D];
  __shared__ int sidx[WPB][MROWS];

  const int lane = threadIdx.x & 31;
  const int wave = threadIdx.x >> 5;
  const int rowbase = (blockIdx.x * WPB + wave) * MROWS;

  // Stage the 32 x D tile of x into LDS (reused K/16 times + loss phase).
  for (int i = lane; i < MROWS * D; i += 32) {
    const int r = i / D, d = i % D;
    xs[wave][r][d] = x[(size_t)(rowbase + r) * D + d];
  }
  __syncthreads();

  const int m = lane & 15;            // A: row in tile / B: column in tile
  const int koff = (lane >> 4) << 1;  // K sub-offset for upper half-wave

  float minv0[8], minv1[8];
  int mini0[8], mini1[8];
#pragma unroll
  for (int r = 0; r < 8; ++r) {
    minv0[r] = 3.4e38f;
    minv1[r] = 3.4e38f;
    mini0[r] = 0;
    mini1[r] = 0;
  }

  for (int kt = 0; kt < K; kt += 16) {
    v8f c0, c1;
#pragma unroll
    for (int r = 0; r < 8; ++r) {
      c0[r] = 0.0f;
      c1[r] = 0.0f;
    }

    const float* wcol = W + (size_t)(kt + m) * D;  // column (kt+m) of W^T
#pragma unroll 4
    for (int d0 = 0; d0 < D; d0 += 4) {
      v2f a0, a1, b;
      b.x = wcol[d0 + koff + 0];
      b.y = wcol[d0 + koff + 1];
      a0.x = xs[wave][m][d0 + koff + 0];
      a0.y = xs[wave][m][d0 + koff + 1];
      a1.x = xs[wave][16 + m][d0 + koff + 0];
      a1.y = xs[wave][16 + m][d0 + koff + 1];
      // Exact f32 matrix ops; two row tiles amortize each B fragment.
      c0 = __builtin_amdgcn_wmma_f32_16x16x4_f32(false, a0, false, b, (short)0,
                                                 c0, false, false);
      c1 = __builtin_amdgcn_wmma_f32_16x16x4_f32(false, a1, false, b, (short)0,
                                                 c1, false, false);
    }

    const float wn = wnorm[kt + m];  // precomputed ||w||^2 (L1-hot)
    const int col = kt + m;
#pragma unroll
    for (int r = 0; r < 8; ++r) {
      const float s0 = wn - 2.0f * c0[r];  // ||x||^2 const per row: dropped
      if (s0 < minv0[r]) {
        minv0[r] = s0;
        mini0[r] = col;
      }
      const float s1 = wn - 2.0f * c1[r];
      if (s1 < minv1[r]) {
        minv1[r] = s1;
        mini1[r] = col;
      }
    }
  }

  // Per-row argmin across the 16 lanes of each half (ties -> lowest index).
#pragma unroll
  for (int r = 0; r < 8; ++r) {
#pragma unroll
    for (int mask = 1; mask <= 8; mask <<= 1) {
      const float ov0 = __shfl_xor(minv0[r], mask);
      const int oi0 = __shfl_xor(mini0[r], mask);
      if (ov0 < minv0[r] || (ov0 == minv0[r] && oi0 < mini0[r])) {
        minv0[r] = ov0;
        mini0[r] = oi0;
      }
      const float ov1 = __shfl_xor(minv1[r], mask);
      const int oi1 = __shfl_xor(mini1[r], mask);
      if (ov1 < minv1[r] || (ov1 == minv1[r] && oi1 < mini1[r])) {
        minv1[r] = ov1;
        mini1[r] = oi1;
      }
    }
  }
  // Tile0 covers rows 0..15, tile1 rows 16..31; within a tile, lanes 0-15
  // hold rows slot..slot (r -> row r), lanes 16-31 hold rows 8+r.
  if (lane == 0) {
#pragma unroll
    for (int r = 0; r < 8; ++r) {
      sidx[wave][r] = mini0[r];
      idx_out[rowbase + r] = (float)mini0[r];
      sidx[wave][16 + r] = mini1[r];
      idx_out[rowbase + 16 + r] = (float)mini1[r];
    }
  } else if (lane == 16) {
#pragma unroll
    for (int r = 0; r < 8; ++r) {
      sidx[wave][8 + r] = mini0[r];
      idx_out[rowbase + 8 + r] = (float)mini0[r];
      sidx[wave][24 + r] = mini1[r];
      idx_out[rowbase + 24 + r] = (float)mini1[r];
    }
  }
  __syncthreads();

  // Gather q = W[idx]; accumulate sum((q - x)^2) for the commitment loss.
  float lsum = 0.0f;
  for (int r = 0; r < MROWS; ++r) {
    const int ci = sidx[wave][r];
    const float* wrow = W + (size_t)ci * D;
    float* qrow = q_out + (size_t)(rowbase + r) * D;
    for (int d = lane; d < D; d += 32) {
      const float qv = wrow[d];
      const float dv = qv - xs[wave][r][d];
      qrow[d] = qv;
      lsum += dv * dv;
    }
  }
#pragma unroll
  for (int mask = 1; mask < 32; mask <<= 1) lsum += __shfl_xor(lsum, mask);
  if (lane == 0)
    atomicAdd(loss_out, lsum * (1.0f / ((float)N * (float)D)));
}

extern "C" void kernel_launch(void* const* d_in, const int* in_sizes, int n_in,
                              void* d_out, int out_size, void* d_ws,
                              size_t ws_size, hipStream_t stream) {
  (void)in_sizes; (void)n_in; (void)out_size; (void)ws_size;
  // setup_inputs() dict order: x_hand, W_hand, x_loc, W_loc, x_ori, W_ori,
  //                            x_mov, W_mov, x_nmf, W_nmf
  const float* x_hand = (const float*)d_in[0];
  const float* W_hand = (const float*)d_in[1];
  const float* x_loc = (const float*)d_in[2];
  const float* W_loc = (const float*)d_in[3];
  const float* x_ori = (const float*)d_in[4];
  const float* W_ori = (const float*)d_in[5];
  const float* x_mov = (const float*)d_in[6];
  const float* W_mov = (const float*)d_in[7];
  const float* x_nmf = (const float*)d_in[8];
  const float* W_nmf = (const float*)d_in[9];

  float* out = (float*)d_out;
  const int N = 16 * 2048;

  // Flat output layout in tuple return order.
  const size_t qh = 0;
  const size_t ql = qh + (size_t)N * 256;
  const size_t qo = ql + (size_t)N * 128;
  const size_t qm = qo + (size_t)N * 64;
  const size_t qn = qm + (size_t)N * 128;
  const size_t ih = qn + (size_t)N * 64;
  const size_t il = ih + (size_t)N;
  const size_t io = il + (size_t)N;
  const size_t im = io + (size_t)N;
  const size_t in_ = im + (size_t)N;
  const size_t lo = in_ + (size_t)N;

  // Workspace: precomputed codeword norms (1280 floats total).
  float* nrm = (float*)d_ws;
  float* n_hand = nrm + 0;     // 512
  float* n_loc = nrm + 512;    // 256
  float* n_ori = nrm + 768;    // 128
  float* n_mov = nrm + 896;    // 256
  float* n_nmf = nrm + 1152;   // 128

  zero_loss_kernel<<<1, 1, 0, stream>>>(out + lo);
  wnorm_kernel<<<512, 32, 0, stream>>>(W_hand, n_hand, 256);
  wnorm_kernel<<<256, 32, 0, stream>>>(W_loc, n_loc, 128);
  wnorm_kernel<<<128, 32, 0, stream>>>(W_ori, n_ori, 64);
  wnorm_kernel<<<256, 32, 0, stream>>>(W_mov, n_mov, 128);
  wnorm_kernel<<<128, 32, 0, stream>>>(W_nmf, n_nmf, 64);

  const dim3 blk(32 * WPB);
  const dim3 grd(N / (MROWS * WPB));
  vq_argmin_gather_kernel<256><<<grd, blk, 0, stream>>>(
      x_hand, W_hand, n_hand, out + qh, out + ih, out + lo, 512, N);
  vq_argmin_gather_kernel<128><<<grd, blk, 0, stream>>>(
      x_loc, W_loc, n_loc, out + ql, out + il, out + lo, 256, N);
  vq_argmin_gather_kernel<64><<<grd, blk, 0, stream>>>(
      x_ori, W_ori, n_ori, out + qo, out + io, out + lo, 128, N);
  vq_argmin_gather_kernel<128><<<grd, blk, 0, stream>>>(
      x_mov, W_mov, n_mov, out + qm, out + im, out + lo, 256, N);
  vq_argmin_gather_kernel<64><<<grd, blk, 0, stream>>>(
      x_nmf, W_nmf, n_nmf, out + qn, out + in_, out + lo, 128, N);
}